// SplineConv_4363686772849
// MI455X (gfx1250) — compile-verified
//
#include <hip/hip_runtime.h>

typedef __attribute__((ext_vector_type(2))) float v2f;
typedef __attribute__((ext_vector_type(8))) float v8f;

#define N_NODES 50000
#define N_EDGES 1600000
#define F 128          // F_IN == F_OUT
#define NCAT 384       // [W0 | W1 | Wroot]

// ---------------------------------------------------------------------------
// Kernel 1: zero accumulators (acc[50000*128] followed contiguously by cnt[50000])
// ---------------------------------------------------------------------------
__global__ void zero_kernel(float* __restrict__ p, int n) {
    int i = blockIdx.x * blockDim.x + threadIdx.x;
    if (i < n) p[i] = 0.0f;
}

// ---------------------------------------------------------------------------
// Kernel 2: WMMA f32 GEMM  XW = x(50000x128) @ Wcat(128x384)
//   grid (3125, 3), block 256 = 8 waves; wave w of block handles
//   mtile = blockIdx.x, ntile = blockIdx.y*8 + w.  50000 == 3125*16 exactly,
//   so every lane is always active (EXEC all-1s, as WMMA requires).
// A-frag (16x4 f32, 2 VGPR): lanes 0-15 -> M=lane, K={k0,k0+1};
//                            lanes 16-31 -> M=lane-16, K={k0+2,k0+3}
// B-frag (4x16 f32, 2 VGPR): lanes 0-15 -> N=lane, K={k0,k0+1};
//                            lanes 16-31 -> N=lane-16, K={k0+2,k0+3}
// C/D (16x16 f32, 8 VGPR):   vgpr i, lanes 0-15 -> (M=i, N=lane);
//                            lanes 16-31 -> (M=i+8, N=lane-16)
// Columns 0-127 -> xw0, 128-255 -> xw1, 256-383 -> x@root (straight to d_out).
// ---------------------------------------------------------------------------
__global__ __launch_bounds__(256)
void gemm_wmma(const float* __restrict__ x,
               const float* __restrict__ w,      // (2,128,128)
               const float* __restrict__ wroot,  // (128,128)
               float* __restrict__ xw0,
               float* __restrict__ xw1,
               float* __restrict__ xroot_out) {
    const int lane  = threadIdx.x & 31;
    const int wave  = threadIdx.x >> 5;
    const int mtile = blockIdx.x;                 // 0..3124
    const int ntile = blockIdx.y * 8 + wave;      // 0..23

    const int row0  = mtile * 16;
    const int aRow  = row0 + (lane & 15);
    const int koff  = (lane < 16) ? 0 : 2;        // same half-split for A and B
    const int col   = ntile * 16 + (lane & 15);   // global output column 0..383

    // Select which weight matrix provides this B column.
    const float* bsrc;
    if (col < F)            bsrc = w + col;                    // W0
    else if (col < 2 * F)   bsrc = w + F * F + (col - F);      // W1
    else                    bsrc = wroot + (col - 2 * F);      // Wroot

    v8f c = {};
    const float* aBase = x + (size_t)aRow * F + koff;
#pragma unroll
    for (int k0 = 0; k0 < F; k0 += 4) {
        v2f a, b;
        a.x = aBase[k0];
        a.y = aBase[k0 + 1];
        const float* bp = bsrc + (size_t)(k0 + koff) * F;
        b.x = bp[0];
        b.y = bp[F];
        // 8 args: (neg_a, A, neg_b, B, c_mod, C, reuse_a, reuse_b)
        c = __builtin_amdgcn_wmma_f32_16x16x4_f32(
                false, a, false, b, (short)0, c, false, false);
    }

    // Write back: vgpr i holds row (row0 + i) or (row0 + 8 + i), column `col`.
    const int srow = row0 + ((lane < 16) ? 0 : 8);
    float* dstbase;
    if (col < F)            dstbase = xw0 + col;
    else if (col < 2 * F)   dstbase = xw1 + (col - F);
    else                    dstbase = xroot_out + (col - 2 * F);
#pragma unroll
    for (int i = 0; i < 8; ++i) {
        dstbase[(size_t)(srow + i) * F] = c[i];
    }
}

// ---------------------------------------------------------------------------
// Kernel 3: edge gather + lerp + atomic scatter.
// One wave32 per edge; each lane handles 4 consecutive features (float4).
// xw rows (512 B) live in L2 (51 MB total << 192 MB), so gathers are L2 hits.
// ---------------------------------------------------------------------------
__global__ __launch_bounds__(256)
void edge_scatter(const float* __restrict__ xw0,
                  const float* __restrict__ xw1,
                  const float* __restrict__ edge_attr,
                  const long long* __restrict__ edge_index, // int64, (2, E)
                  float* __restrict__ acc,
                  float* __restrict__ cnt) {
    const int lane = threadIdx.x & 31;
    const long long e = (long long)blockIdx.x * 8 + (threadIdx.x >> 5);
    if (e >= N_EDGES) return;

    const int src = (int)edge_index[e];
    const int dst = (int)edge_index[(long long)N_EDGES + e];
    const float u  = edge_attr[e];
    const float um = 1.0f - u;

    const float4 a = ((const float4*)(xw0 + (size_t)src * F))[lane];
    const float4 b = ((const float4*)(xw1 + (size_t)src * F))[lane];

    float4 m;
    m.x = um * a.x + u * b.x;
    m.y = um * a.y + u * b.y;
    m.z = um * a.z + u * b.z;
    m.w = um * a.w + u * b.w;

    float* d = acc + (size_t)dst * F + lane * 4;
    atomicAdd(d + 0, m.x);
    atomicAdd(d + 1, m.y);
    atomicAdd(d + 2, m.z);
    atomicAdd(d + 3, m.w);

    if (lane == 0) atomicAdd(cnt + dst, 1.0f);
}

// ---------------------------------------------------------------------------
// Kernel 4: out = xroot (already in d_out) + acc / max(cnt, 1)
// ---------------------------------------------------------------------------
__global__ void finalize_kernel(const float* __restrict__ acc,
                                const float* __restrict__ cnt,
                                float* __restrict__ out) {
    int i = blockIdx.x * blockDim.x + threadIdx.x;
    if (i < N_NODES * F) {
        float cdeg = fmaxf(cnt[i >> 7], 1.0f);   // i / 128
        out[i] = out[i] + acc[i] / cdeg;
    }
}

// ---------------------------------------------------------------------------
extern "C" void kernel_launch(void* const* d_in, const int* in_sizes, int n_in,
                              void* d_out, int out_size, void* d_ws, size_t ws_size,
                              hipStream_t stream) {
    const float*     x           = (const float*)d_in[0];
    const float*     edge_attr   = (const float*)d_in[1];
    const float*     weight      = (const float*)d_in[2];
    const float*     root_weight = (const float*)d_in[3];
    const long long* edge_index  = (const long long*)d_in[4];
    float*           out         = (float*)d_out;

    // Workspace layout (floats): xw0 | xw1 | acc | cnt   (~77 MB)
    float* ws  = (float*)d_ws;
    float* xw0 = ws;
    float* xw1 = xw0 + (size_t)N_NODES * F;
    float* acc = xw1 + (size_t)N_NODES * F;
    float* cnt = acc + (size_t)N_NODES * F;   // contiguous after acc

    // 1) zero acc + cnt (must happen every call: graph replays reuse buffers)
    const int nz = N_NODES * F + N_NODES;
    zero_kernel<<<(nz + 255) / 256, 256, 0, stream>>>(acc, nz);

    // 2) fused GEMM: xw0, xw1 -> ws ; x@root_weight -> d_out
    gemm_wmma<<<dim3(3125, 3), 256, 0, stream>>>(x, weight, root_weight,
                                                 xw0, xw1, out);

    // 3) per-edge gather / lerp / scatter-add (wave per edge)
    edge_scatter<<<(N_EDGES + 7) / 8, 256, 0, stream>>>(xw0, xw1, edge_attr,
                                                        edge_index, acc, cnt);

    // 4) mean-normalize and add root term
    finalize_kernel<<<(N_NODES * F + 255) / 256, 256, 0, stream>>>(acc, cnt, out);
}